// TimeConv_17652315587472
// MI455X (gfx1250) — compile-verified
//
#include <hip/hip_runtime.h>
#include <math.h>

typedef float v2f __attribute__((ext_vector_type(2)));
typedef float v4f __attribute__((ext_vector_type(4)));
typedef float v8f __attribute__((ext_vector_type(8)));

#define B_    64
#define C_    8
#define FC_   16
#define WINS_ 30
#define TWIN_ 512
#define SE_   8
#define ALPHA_ 0.01f

#define NWIN (B_ * C_ * FC_ * WINS_)   // 245760 windows of 512 samples
#define ROWS (B_ * C_)                 // 512 (b,c) rows

// ---------------------------------------------------------------------------
// Phase 1: streaming per-window mean + max. One wave32 per 2KB window.
// Bandwidth-bound: 503MB single pass, non-temporal (dataset >> 192MB L2).
// ---------------------------------------------------------------------------
__global__ __launch_bounds__(256) void win_reduce(const float* __restrict__ x,
                                                  float* __restrict__ meanArr,
                                                  float* __restrict__ maxArr) {
    const int lane = threadIdx.x & 31;
    const int wave = threadIdx.x >> 5;
    const int n = blockIdx.x * 8 + wave;          // window id, < NWIN by grid
    const v4f* p = (const v4f*)(x + (size_t)n * TWIN_);

    float s = 0.0f;
    float m = -INFINITY;
#pragma unroll
    for (int j = 0; j < 4; ++j) {
        v4f v = __builtin_nontemporal_load(&p[j * 32 + lane]);  // b128, coalesced
        s += (v.x + v.y) + (v.z + v.w);
        m = fmaxf(m, fmaxf(fmaxf(v.x, v.y), fmaxf(v.z, v.w)));
    }
#pragma unroll
    for (int off = 16; off >= 1; off >>= 1) {
        s += __shfl_xor(s, off, 32);
        m = fmaxf(m, __shfl_xor(m, off, 32));
    }
    if (lane == 0) {
        meanArr[n] = s * (1.0f / TWIN_);
        maxArr[n]  = m;
    }
}

// ---------------------------------------------------------------------------
// Phase 2: variance over FC + SE MLP via V_WMMA_F32_16X16X4_F32.
// One full wave32 per block (EXEC all-ones as WMMA requires); each wave owns
// 16 (b,c) rows. GEMM1: v(16x30 pad32) @ w1(30x8 pad16), 8 WMMAs.
// GEMM2: h(16x8) @ w2(8x30 padN32), 2 N-tiles x 2 K-chunks = 4 WMMAs.
// ---------------------------------------------------------------------------
__global__ __launch_bounds__(32) void se_gates(const float* __restrict__ meanArr,
                                               const float* __restrict__ w1,
                                               const float* __restrict__ b1,
                                               const float* __restrict__ w2,
                                               const float* __restrict__ b2,
                                               float* __restrict__ gArr) {
    __shared__ float vtile[16][32];   // [row-in-tile][K=win, padded 30->32]
    __shared__ float htile[16][8];    // hidden activations, A-source for GEMM2

    const int lane = threadIdx.x;     // 0..31, all lanes active
    const int r0 = blockIdx.x * 16;   // first (b,c) row of this tile

    // zero K-padding (cols 30,31)
    for (int i = lane; i < 16 * 32; i += 32) (&vtile[0][0])[i] = 0.0f;
    __syncthreads();

    // variance across FC for each (row, window)
    for (int pidx = lane; pidx < 16 * WINS_; pidx += 32) {
        const int rr = pidx / WINS_;
        const int w  = pidx - rr * WINS_;
        const int row = r0 + rr;
        float s1 = 0.0f, s2 = 0.0f;
#pragma unroll
        for (int f = 0; f < FC_; ++f) {
            const float mv = meanArr[(row * FC_ + f) * WINS_ + w];
            s1 += mv;
            s2 += mv * mv;
        }
        const float mu = s1 * (1.0f / FC_);
        vtile[rr][w] = s2 * (1.0f / FC_) - mu * mu;
    }
    __syncthreads();

    // ISA 7.12.2 fragment indexing (wave32, f32):
    // A 16x4: lanes 0-15 hold M=lane, VGPR0=K+0/VGPR1=K+1; lanes 16-31: K+2/K+3
    // B 4x16: lanes 0-15 hold N=lane, VGPR0=K+0/VGPR1=K+1; lanes 16-31: K+2/K+3
    // C/D 16x16: VGPR r -> (M = r + (lane>=16 ? 8:0), N = lane&15)
    const int mrow  = lane & 15;
    const int khalf = (lane < 16) ? 0 : 2;
    const int ncol  = lane & 15;
    const int mD    = (lane < 16) ? 0 : 8;

    // ---- GEMM1: h_pre = v @ w1, K = 0..31 (30 valid), N = 0..15 (8 valid) ---
    v8f acc = {0.f, 0.f, 0.f, 0.f, 0.f, 0.f, 0.f, 0.f};
#pragma unroll
    for (int kb = 0; kb < 32; kb += 4) {
        const int k0 = kb + khalf;
        const int k1 = k0 + 1;
        v2f a, b;
        a.x = vtile[mrow][k0];
        a.y = vtile[mrow][k1];
        // w1 is (WINS_, SE_); predicate via multiply (no EXEC-touching branches)
        const float pn = (ncol < SE_) ? 1.0f : 0.0f;
        const int nc = ncol < SE_ ? ncol : (SE_ - 1);
        const int kk0 = k0 < WINS_ ? k0 : (WINS_ - 1);
        const int kk1 = k1 < WINS_ ? k1 : (WINS_ - 1);
        b.x = w1[kk0 * SE_ + nc] * ((k0 < WINS_) ? pn : 0.0f);
        b.y = w1[kk1 * SE_ + nc] * ((k1 < WINS_) ? pn : 0.0f);
        acc = __builtin_amdgcn_wmma_f32_16x16x4_f32(false, a, false, b,
                                                    (short)0, acc, false, false);
    }

    // bias + leaky_relu; scatter D-layout into A-layout staging (htile)
    const float b1v = b1[(ncol < SE_) ? ncol : (SE_ - 1)];
#pragma unroll
    for (int r = 0; r < 8; ++r) {
        float hv = acc[r] + b1v;
        hv = (hv > 0.0f) ? hv : ALPHA_ * hv;
        if (ncol < SE_) htile[mD + r][ncol] = hv;
    }
    __syncthreads();

    // ---- GEMM2: g_pre = h @ w2, K = 0..7, N-tile0 = 0..15, N-tile1 = 16..31 --
    v8f acc0 = {0.f, 0.f, 0.f, 0.f, 0.f, 0.f, 0.f, 0.f};
    v8f acc1 = {0.f, 0.f, 0.f, 0.f, 0.f, 0.f, 0.f, 0.f};
#pragma unroll
    for (int kb = 0; kb < 8; kb += 4) {
        const int k0 = kb + khalf;      // always < 8
        const int k1 = k0 + 1;
        v2f a, bt0, bt1;
        a.x = htile[mrow][k0];
        a.y = htile[mrow][k1];
        // w2 is (SE_, WINS_)
        bt0.x = w2[k0 * WINS_ + ncol];  // ncol < 16 < 30: always valid
        bt0.y = w2[k1 * WINS_ + ncol];
        const int n1 = 16 + ncol;
        const float p1 = (n1 < WINS_) ? 1.0f : 0.0f;
        const int nn1 = (n1 < WINS_) ? n1 : (WINS_ - 1);
        bt1.x = w2[k0 * WINS_ + nn1] * p1;
        bt1.y = w2[k1 * WINS_ + nn1] * p1;
        acc0 = __builtin_amdgcn_wmma_f32_16x16x4_f32(false, a, false, bt0,
                                                     (short)0, acc0, false, false);
        acc1 = __builtin_amdgcn_wmma_f32_16x16x4_f32(false, a, false, bt1,
                                                     (short)0, acc1, false, false);
    }

    // bias + sigmoid, write gates (B*C, WINS)
    const float b2a = b2[ncol];                                   // w = ncol < 30
    const int n1 = 16 + ncol;
    const float b2b = b2[(n1 < WINS_) ? n1 : (WINS_ - 1)];
#pragma unroll
    for (int r = 0; r < 8; ++r) {
        const int row = r0 + mD + r;
        const float g0 = 1.0f / (1.0f + __expf(-(acc0[r] + b2a)));
        gArr[row * WINS_ + ncol] = g0;
        if (n1 < WINS_) {
            const float g1 = 1.0f / (1.0f + __expf(-(acc1[r] + b2b)));
            gArr[row * WINS_ + n1] = g1;
        }
    }
}

// ---------------------------------------------------------------------------
// Phase 3: out[b,c,f] = sum_w mx[b,c,f,w] * g[b,c,w]   (8192 outputs, trivial)
// ---------------------------------------------------------------------------
__global__ __launch_bounds__(256) void gate_apply(const float* __restrict__ mxArr,
                                                  const float* __restrict__ gArr,
                                                  float* __restrict__ out) {
    const int i = blockIdx.x * 256 + threadIdx.x;   // (b,c,f) flat
    if (i >= B_ * C_ * FC_) return;
    const int row = i / FC_;                        // (b,c) flat
    const float* mx = mxArr + (size_t)i * WINS_;
    const float* g  = gArr + (size_t)row * WINS_;
    float s = 0.0f;
#pragma unroll
    for (int w = 0; w < WINS_; ++w) s = fmaf(mx[w], g[w], s);
    out[i] = s;
}

// ---------------------------------------------------------------------------
extern "C" void kernel_launch(void* const* d_in, const int* in_sizes, int n_in,
                              void* d_out, int out_size, void* d_ws, size_t ws_size,
                              hipStream_t stream) {
    const float* x  = (const float*)d_in[0];
    const float* w1 = (const float*)d_in[1];
    const float* b1 = (const float*)d_in[2];
    const float* w2 = (const float*)d_in[3];
    const float* b2 = (const float*)d_in[4];
    float* out = (float*)d_out;

    float* meanArr = (float*)d_ws;         // NWIN floats
    float* maxArr  = meanArr + NWIN;       // NWIN floats
    float* gArr    = maxArr + NWIN;        // ROWS*WINS_ floats  (~2MB total)

    win_reduce<<<NWIN / 8, 256, 0, stream>>>(x, meanArr, maxArr);
    se_gates<<<ROWS / 16, 32, 0, stream>>>(meanArr, w1, b1, w2, b2, gArr);
    gate_apply<<<(B_ * C_ * FC_ + 255) / 256, 256, 0, stream>>>(maxArr, gArr, out);
}